// LightGCN_xij_item_personal_matrix_833223655947
// MI455X (gfx1250) — compile-verified
//
#include <hip/hip_runtime.h>
#include <math.h>

#define N_USERS 100000
#define N_ITEMS 50000
#define NTOT    (N_USERS + N_ITEMS)
#define D       64
#define LAYERS  3
#define HX      0.5f

typedef float v2f __attribute__((ext_vector_type(2)));
typedef float v8f __attribute__((ext_vector_type(8)));

// ---------------------------------------------------------------------------
// Stage 0: e0 = acc = concat(emb_user, emb_item)
// ---------------------------------------------------------------------------
__global__ void k_init(const float* __restrict__ eu, const float* __restrict__ ei,
                       float* __restrict__ acc, float* __restrict__ ecur) {
    size_t i = (size_t)blockIdx.x * blockDim.x + threadIdx.x;
    const size_t tot = (size_t)NTOT * D;
    if (i >= tot) return;
    float v = (i < (size_t)N_USERS * D) ? eu[i] : ei[i - (size_t)N_USERS * D];
    acc[i]  = v;
    ecur[i] = v;
}

// ---------------------------------------------------------------------------
// Zero a buffer (float4 granularity)
// ---------------------------------------------------------------------------
__global__ void k_zero(float4* __restrict__ p, size_t n4) {
    size_t i = (size_t)blockIdx.x * blockDim.x + threadIdx.x;
    if (i < n4) p[i] = make_float4(0.f, 0.f, 0.f, 0.f);
}

// ---------------------------------------------------------------------------
// SpMM scatter: one wave (32 lanes) per nonzero; each lane handles 2 floats of
// the 64-wide row -> 256B coalesced gather + coalesced f32 atomic scatter.
// Both streams are L2-resident (feature matrix = 38.4MB < 192MB L2).
// ---------------------------------------------------------------------------
__global__ void k_spmm(const int* __restrict__ row, const int* __restrict__ col,
                       const float* __restrict__ val, const float* __restrict__ x,
                       float* __restrict__ y, int nnz) {
    int g = blockIdx.x * (blockDim.x >> 5) + (threadIdx.x >> 5);
    if (g >= nnz) return;
    const int lane = threadIdx.x & 31;
    const int c = col[g];
    const int r = row[g];
    const float v = val[g];
    const float2 xv = *reinterpret_cast<const float2*>(x + (size_t)c * D + lane * 2);
    float* yp = y + (size_t)r * D + lane * 2;
    atomicAdd(yp,     v * xv.x);
    atomicAdd(yp + 1, v * xv.y);
}

// ---------------------------------------------------------------------------
// acc += e (float4 granularity)
// ---------------------------------------------------------------------------
__global__ void k_accum(float4* __restrict__ acc, const float4* __restrict__ e, size_t n4) {
    size_t i = (size_t)blockIdx.x * blockDim.x + threadIdx.x;
    if (i >= n4) return;
    float4 a = acc[i];
    float4 b = e[i];
    a.x += b.x; a.y += b.y; a.z += b.z; a.w += b.w;
    acc[i] = a;
}

// ---------------------------------------------------------------------------
// Final fused head: gather 16 rows -> WMMA fp32 GEMM (ue@Wu^T, ie@Wi^T)
// -> softmax / sigmoid / weighted dot -> out[B].
// 128 threads = 4 wave32; wave w owns output columns [16w, 16w+16).
// ---------------------------------------------------------------------------
__global__ __launch_bounds__(128)
void k_final(const float* __restrict__ acc,
             const float* __restrict__ w_user, const float* __restrict__ w_item,
             const float* __restrict__ xe1, const float* __restrict__ xe0,
             const int* __restrict__ users, const int* __restrict__ items,
             const int* __restrict__ xij, float* __restrict__ out, int B) {
    __shared__ float ue[16][D];
    __shared__ float ie[16][D];
    __shared__ float uo[16][D + 2];
    __shared__ float io[16][D + 2];
    __shared__ int s_users[16];
    __shared__ int s_items[16];
    __shared__ int s_xij[16];

    const int tid = threadIdx.x;
    const int r0  = blockIdx.x * 16;

    if (tid < 16) {
        const int r = r0 + tid;
        s_users[tid] = (r < B) ? users[r] : 0;
        s_items[tid] = (r < B) ? items[r] : 0;
        s_xij[tid]   = (r < B) ? xij[r]   : 0;
    }
    __syncthreads();

    // Gather the two 16x64 tiles, folding the LightGCN 1/(LAYERS+1) scale.
    const float inv = 1.0f / (float)(LAYERS + 1);
    for (int idx = tid; idx < 16 * D; idx += 128) {
        const int r = idx >> 6;
        const int k = idx & (D - 1);
        ue[r][k] = acc[(size_t)s_users[r] * D + k] * inv;
        ie[r][k] = acc[((size_t)N_USERS + (size_t)s_items[r]) * D + k] * inv;
    }
    __syncthreads();

    // ---- WMMA fp32 16x16x4: out[m][n] = sum_k A[m][k] * W[n][k] ----
    const int wave  = tid >> 5;
    const int lane  = tid & 31;
    const int n0    = wave * 16;
    const int m     = lane & 15;          // A row / C column index
    const int khalf = (lane < 16) ? 0 : 2; // K sub-slot per ISA A/B layout
    const int ncol  = n0 + m;             // B column (= W row, since B = W^T)

    v8f cu = {};
    v8f ci = {};
#pragma unroll
    for (int kk = 0; kk < D; kk += 4) {
        v2f a, b;
        a.x = ue[m][kk + khalf];
        a.y = ue[m][kk + khalf + 1];
        b.x = w_user[(size_t)ncol * D + kk + khalf];
        b.y = w_user[(size_t)ncol * D + kk + khalf + 1];
        cu = __builtin_amdgcn_wmma_f32_16x16x4_f32(false, a, false, b, (short)0, cu,
                                                   false, false);
        v2f a2, b2;
        a2.x = ie[m][kk + khalf];
        a2.y = ie[m][kk + khalf + 1];
        b2.x = w_item[(size_t)ncol * D + kk + khalf];
        b2.y = w_item[(size_t)ncol * D + kk + khalf + 1];
        ci = __builtin_amdgcn_wmma_f32_16x16x4_f32(false, a2, false, b2, (short)0, ci,
                                                   false, false);
    }

    // C/D layout: VGPR j -> row j (lanes 0-15) / row j+8 (lanes 16-31), N = lane&15
#pragma unroll
    for (int j = 0; j < 8; ++j) {
        const int rr = j + ((lane < 16) ? 0 : 8);
        uo[rr][ncol] = cu[j];
        io[rr][ncol] = ci[j];
    }
    __syncthreads();

    // ---- per-row softmax(ue), sigmoid(ie), dot; 8 lanes per row ----
    const int r   = tid >> 3;  // 0..15
    const int sub = tid & 7;   // 0..7  (8 contiguous lanes within a wave32)

    float mx = -1e30f;
#pragma unroll
    for (int j = 0; j < 8; ++j) mx = fmaxf(mx, uo[r][sub * 8 + j]);
    mx = fmaxf(mx, __shfl_xor(mx, 1, 32));
    mx = fmaxf(mx, __shfl_xor(mx, 2, 32));
    mx = fmaxf(mx, __shfl_xor(mx, 4, 32));

    float s = 0.f;
#pragma unroll
    for (int j = 0; j < 8; ++j) s += expf(uo[r][sub * 8 + j] - mx);
    s += __shfl_xor(s, 1, 32);
    s += __shfl_xor(s, 2, 32);
    s += __shfl_xor(s, 4, 32);

    const float coef = (1.0f - HX) / s;
    float dot = 0.f;
#pragma unroll
    for (int j = 0; j < 8; ++j) {
        const int k = sub * 8 + j;
        const float p  = expf(uo[r][k] - mx) * coef;
        const float sg = 1.0f / (1.0f + expf(-io[r][k]));
        dot += p * sg;
    }
    dot += __shfl_xor(dot, 1, 32);
    dot += __shfl_xor(dot, 2, 32);
    dot += __shfl_xor(dot, 4, 32);

    if (sub == 0 && (r0 + r) < B) {
        const float xsel = (s_xij[r] > 0) ? xe1[s_items[r]] : xe0[s_items[r]];
        const float sg   = 1.0f / (1.0f + expf(-xsel));
        out[r0 + r] = dot + HX * sg;
    }
}

// ---------------------------------------------------------------------------
extern "C" void kernel_launch(void* const* d_in, const int* in_sizes, int n_in,
                              void* d_out, int out_size, void* d_ws, size_t ws_size,
                              hipStream_t stream) {
    const float* emb_user = (const float*)d_in[0];
    const float* emb_item = (const float*)d_in[1];
    const float* w_user   = (const float*)d_in[2];
    const float* w_item   = (const float*)d_in[3];
    const float* xe1      = (const float*)d_in[4];
    const float* xe0      = (const float*)d_in[5];
    const float* g_val    = (const float*)d_in[6];
    const int*   g_row    = (const int*)d_in[7];
    const int*   g_col    = (const int*)d_in[8];
    const int*   users    = (const int*)d_in[9];
    const int*   items    = (const int*)d_in[10];
    const int*   xij      = (const int*)d_in[11];
    const int    nnz      = in_sizes[6];
    const int    B        = in_sizes[9];

    const size_t nd = (size_t)NTOT * D;     // 9.6M floats
    float* acc = (float*)d_ws;              // 38.4 MB
    float* e0  = acc + nd;                  // 38.4 MB
    float* e1  = e0 + nd;                   // 38.4 MB

    const int t = 256;
    k_init<<<(unsigned)((nd + t - 1) / t), t, 0, stream>>>(emb_user, emb_item, acc, e0);

    float* ecur  = e0;
    float* enext = e1;
    const size_t n4 = nd / 4;
    const int wavesPerBlock = t / 32;
    for (int l = 0; l < LAYERS; ++l) {
        k_zero<<<(unsigned)((n4 + t - 1) / t), t, 0, stream>>>((float4*)enext, n4);
        k_spmm<<<(nnz + wavesPerBlock - 1) / wavesPerBlock, t, 0, stream>>>(
            g_row, g_col, g_val, ecur, enext, nnz);
        k_accum<<<(unsigned)((n4 + t - 1) / t), t, 0, stream>>>(
            (float4*)acc, (const float4*)enext, n4);
        float* tmp = ecur; ecur = enext; enext = tmp;
    }

    k_final<<<(B + 15) / 16, 128, 0, stream>>>(acc, w_user, w_item, xe1, xe0,
                                               users, items, xij, (float*)d_out, B);
}